// PScanTriton_27393301414039
// MI455X (gfx1250) — compile-verified
//
#include <hip/hip_runtime.h>

// ---------------------------------------------------------------------------
// Complex linear-recurrence scan  y_t = a_t * y_{t-1} + x_t  (B=8, L=4096, D=256)
// Pure streaming problem: ~192 MiB minimum traffic -> ~8us at 23.3 TB/s.
// 3-phase chunked scan, streaming inputs through LDS with CDNA5 async
// global->LDS copies (ASYNCcnt) and double buffering.
// ---------------------------------------------------------------------------

#define BDIM 256

constexpr int Bn    = 8;
constexpr int Ln    = 4096;
constexpr int Dn    = 256;
constexpr int Cn    = 64;            // chunks along L
constexpr int Tn    = Ln / Cn;       // 64 timesteps per chunk
constexpr int STEPn = 8;             // timesteps per staging tile
constexpr int NSTG  = Tn / STEPn;    // 8 staging tiles per chunk
constexpr int TILE  = STEPn * Dn;    // 2048 floats per array per tile (8 KiB)

#if __has_builtin(__builtin_amdgcn_global_load_async_to_lds_b128) && \
    __has_builtin(__builtin_amdgcn_s_wait_asynccnt)
#define USE_ASYNC 1
#else
#define USE_ASYNC 0
#endif

// 16-byte vector type matching the async-LDS builtin's pointee type.
typedef int v4i __attribute__((vector_size(16)));
typedef __attribute__((address_space(1))) v4i* g_v4i_ptr;   // global
typedef __attribute__((address_space(3))) v4i* l_v4i_ptr;   // LDS

// Copy 16 contiguous bytes global -> LDS.
__device__ __forceinline__ void copy16(const float* g, float* l) {
#if USE_ASYNC
    // gfx1250: GLOBAL_LOAD_ASYNC_TO_LDS_B128 (tracked with ASYNCcnt)
    __builtin_amdgcn_global_load_async_to_lds_b128(
        (g_v4i_ptr)g, (l_v4i_ptr)l, /*offset=*/0, /*cpol=*/0);
#else
    *(float4*)l = *(const float4*)g;
#endif
}

__device__ __forceinline__ void wait_async0() {
#if USE_ASYNC
    __builtin_amdgcn_s_wait_asynccnt(0);   // s_wait_asynccnt 0
#endif
}

// Stage one 8x256 tile of all four input arrays into LDS buffer `buf`.
// Tile data is contiguous in memory ([t,d] rows, stride D) => 2048 floats/array.
__device__ __forceinline__ void stage_tile(const float* __restrict__ Ar,
                                           const float* __restrict__ Ai,
                                           const float* __restrict__ Xr,
                                           const float* __restrict__ Xi,
                                           size_t goff,
                                           float (*sm)[TILE],   // sm[4][TILE]
                                           int tid) {
    const int e0 = tid * 4;
    const int e1 = e0 + (TILE / 2);
    copy16(Ar + goff + e0, &sm[0][e0]);
    copy16(Ar + goff + e1, &sm[0][e1]);
    copy16(Ai + goff + e0, &sm[1][e0]);
    copy16(Ai + goff + e1, &sm[1][e1]);
    copy16(Xr + goff + e0, &sm[2][e0]);
    copy16(Xr + goff + e1, &sm[2][e1]);
    copy16(Xi + goff + e0, &sm[3][e0]);
    copy16(Xi + goff + e1, &sm[3][e1]);
}

// ---------------------------------------------------------------------------
// Phase 1: per-chunk local scan from identity state; emit chunk aggregate
// (P = prod a, Y = local scan value at chunk end) as float4 per (b,c,d).
// ---------------------------------------------------------------------------
__global__ __launch_bounds__(BDIM) void pscan_phase1(
    const float* __restrict__ Ar, const float* __restrict__ Ai,
    const float* __restrict__ Xr, const float* __restrict__ Xi,
    float4* __restrict__ agg) {
    __shared__ float sm[2][4][TILE];   // 64 KiB: double-buffered 4-array tiles

    const int d = threadIdx.x;
    const int c = blockIdx.x;
    const int b = blockIdx.y;
    const size_t base = ((size_t)b * Ln + (size_t)c * Tn) * Dn;

    float pr = 1.0f, pi = 0.0f;        // cumulative product of a over chunk
    float yr = 0.0f, yi = 0.0f;        // local scan state (identity start)

    stage_tile(Ar, Ai, Xr, Xi, base, sm[0], d);

    for (int s = 0; s < NSTG; ++s) {
        const int buf = s & 1;
        wait_async0();        // this wave's tile-s data has landed
        __syncthreads();      // all waves landed tile s; all done reading buf^1
        if (s + 1 < NSTG)
            stage_tile(Ar, Ai, Xr, Xi, base + (size_t)(s + 1) * TILE,
                       sm[buf ^ 1], d);

        const float* sAr = sm[buf][0];
        const float* sAi = sm[buf][1];
        const float* sXr = sm[buf][2];
        const float* sXi = sm[buf][3];
#pragma unroll
        for (int t = 0; t < STEPn; ++t) {
            const int idx = t * Dn + d;
            const float ar = sAr[idx], ai = sAi[idx];
            const float xr = sXr[idx], xi = sXi[idx];
            const float nyr = fmaf(ar, yr, xr) - ai * yi;
            const float nyi = fmaf(ar, yi, xi) + ai * yr;
            const float npr = ar * pr - ai * pi;
            const float npi = ar * pi + ai * pr;
            yr = nyr; yi = nyi; pr = npr; pi = npi;
        }
    }

    agg[(b * Cn + c) * Dn + d] = make_float4(pr, pi, yr, yi);
}

// ---------------------------------------------------------------------------
// Phase 2: per-channel exclusive scan over the Cn chunk aggregates.
// carry[b,c,d] = state y entering chunk c.
// ---------------------------------------------------------------------------
__global__ __launch_bounds__(BDIM) void pscan_phase2(
    const float4* __restrict__ agg, float2* __restrict__ carry) {
    const int d = threadIdx.x;
    const int b = blockIdx.x;
    float yr = 0.0f, yi = 0.0f;
    for (int c = 0; c < Cn; ++c) {
        const int idx = (b * Cn + c) * Dn + d;
        carry[idx] = make_float2(yr, yi);
        const float4 g = agg[idx];     // (pr, pi, y_end_r, y_end_i)
        const float nyr = fmaf(g.x, yr, g.z) - g.y * yi;
        const float nyi = fmaf(g.x, yi, g.w) + g.y * yr;
        yr = nyr; yi = nyi;
    }
}

// ---------------------------------------------------------------------------
// Phase 3: re-run the chunk recurrence seeded with the carry; write final
// interleaved (re,im) output.
// ---------------------------------------------------------------------------
__global__ __launch_bounds__(BDIM) void pscan_phase3(
    const float* __restrict__ Ar, const float* __restrict__ Ai,
    const float* __restrict__ Xr, const float* __restrict__ Xi,
    const float2* __restrict__ carry, float2* __restrict__ out) {
    __shared__ float sm[2][4][TILE];

    const int d = threadIdx.x;
    const int c = blockIdx.x;
    const int b = blockIdx.y;
    const size_t base = ((size_t)b * Ln + (size_t)c * Tn) * Dn;

    const float2 h = carry[(b * Cn + c) * Dn + d];
    float yr = h.x, yi = h.y;

    stage_tile(Ar, Ai, Xr, Xi, base, sm[0], d);

    for (int s = 0; s < NSTG; ++s) {
        const int buf = s & 1;
        wait_async0();
        __syncthreads();
        if (s + 1 < NSTG)
            stage_tile(Ar, Ai, Xr, Xi, base + (size_t)(s + 1) * TILE,
                       sm[buf ^ 1], d);

        const float* sAr = sm[buf][0];
        const float* sAi = sm[buf][1];
        const float* sXr = sm[buf][2];
        const float* sXi = sm[buf][3];
#pragma unroll
        for (int t = 0; t < STEPn; ++t) {
            const int idx = t * Dn + d;
            const float ar = sAr[idx], ai = sAi[idx];
            const float xr = sXr[idx], xi = sXi[idx];
            const float nyr = fmaf(ar, yr, xr) - ai * yi;
            const float nyi = fmaf(ar, yi, xi) + ai * yr;
            yr = nyr; yi = nyi;
            out[base + (size_t)(s * STEPn + t) * Dn + d] = make_float2(yr, yi);
        }
    }
}

// ---------------------------------------------------------------------------
extern "C" void kernel_launch(void* const* d_in, const int* in_sizes, int n_in,
                              void* d_out, int out_size, void* d_ws, size_t ws_size,
                              hipStream_t stream) {
    const float* Ar = (const float*)d_in[0];
    const float* Ai = (const float*)d_in[1];
    const float* Xr = (const float*)d_in[2];
    const float* Xi = (const float*)d_in[3];

    // Scratch layout: [ agg: B*C*D float4 | carry: B*C*D float2 ]  (~3 MiB)
    float4* agg   = (float4*)d_ws;
    float2* carry = (float2*)((char*)d_ws + sizeof(float4) * (size_t)Bn * Cn * Dn);

    dim3 blk(BDIM);
    pscan_phase1<<<dim3(Cn, Bn), blk, 0, stream>>>(Ar, Ai, Xr, Xi, agg);
    pscan_phase2<<<dim3(Bn),     blk, 0, stream>>>(agg, carry);
    pscan_phase3<<<dim3(Cn, Bn), blk, 0, stream>>>(Ar, Ai, Xr, Xi, carry,
                                                   (float2*)d_out);
}